// Attention_17093969838577
// MI455X (gfx1250) — compile-verified
//
#include <hip/hip_runtime.h>
#include <stdint.h>

typedef __attribute__((ext_vector_type(2))) float v2f;
typedef __attribute__((ext_vector_type(8))) float v8f;
typedef __attribute__((ext_vector_type(4))) unsigned int u32x4;
typedef __attribute__((ext_vector_type(8))) int i32x8;
typedef __attribute__((ext_vector_type(4))) int i32x4;

#define NUM_HEADS    32
#define NUM_KV_HEADS 8
#define HEAD_DIM     128
#define BLOCK_SIZE   256
#define MAX_BLOCKS   8
#define GROUP        4
#define WAVES        4

#define ROW_F   132                 // 128 dims + 4 floats TDM pad -> conflict-free LDS banks
#define TILE_F  (16 * ROW_F)        // 2112 floats per 16-token tile
// dynamic LDS layout (floats):
//   K tiles : (w*2 + buf)        * TILE_F      (8 tiles)
//   V tiles : (8 + w*2 + buf)    * TILE_F      (8 tiles)
//   merge   : acc[4][4][128], m[4][4], l[4][4]
#define MERGE_F (16 * TILE_F)
#define LDS_FLOATS (MERGE_F + WAVES * GROUP * HEAD_DIM + 2 * WAVES * GROUP)
#define LDS_BYTES  (LDS_FLOATS * 4)

extern __shared__ float smem[];

// ---- TDM: DMA one 16x128 f32 tile (row stride 1024 elems) into LDS, with
// ---- 4-DWORD padding inserted every 128 DWORDs (row stride becomes 132).
// ---- cpol=1 -> TH=NT: KV stream has zero reuse, keep it out of L2 residency.
__device__ __forceinline__ void tdm_load_tile(const float* gbase, unsigned lds_byte_off)
{
    const unsigned long long ga = (unsigned long long)(uintptr_t)gbase;
    u32x4 g0;
    g0[0] = 1u;                                       // count=1, user descriptor
    g0[1] = lds_byte_off;                             // lds_addr
    g0[2] = (unsigned)(ga & 0xffffffffu);             // global_addr[31:0]
    g0[3] = (unsigned)((ga >> 32) & 0x01ffffffu)      // global_addr[56:32]
            | (2u << 30);                             // type = 2 ("image")
    i32x8 g1;
    g1[0] = (int)((2u << 16)      // data_size = 4 bytes
                | (1u << 20)      // pad_enable
                | (6u << 22)      // pad_interval: 2^(6+1) = 128 DWORDs
                | (3u << 25));    // pad_amount: 3+1 = 4 DWORDs
    g1[1] = (int)(128u << 16);                        // tensor_dim0 = 128 (lo16)
    g1[2] = (int)(16u << 16);                         // tensor_dim0 hi=0 | tensor_dim1 = 16
    g1[3] = (int)(128u << 16);                        // tensor_dim1 hi=0 | tile_dim0 = 128
    g1[4] = (int)16u;                                 // tile_dim1 = 16, tile_dim2 = 0 (2-D)
    g1[5] = (int)(NUM_KV_HEADS * HEAD_DIM);           // tensor_dim0_stride = 1024 (lo32)
    g1[6] = 0;
    g1[7] = 0;
    const i32x4 gz = {0, 0, 0, 0};
#if __clang_major__ >= 23
    const i32x8 gz8 = {0, 0, 0, 0, 0, 0, 0, 0};
    __builtin_amdgcn_tensor_load_to_lds(g0, g1, gz, gz, gz8, 1);
#else
    __builtin_amdgcn_tensor_load_to_lds(g0, g1, gz, gz, 1);
#endif
}

__device__ __forceinline__ void prefetch_chunk(const float* kc, const float* vc,
                                               const int* bt_row, int kh, int c,
                                               int w, int buf)
{
    const int  cc   = c * 16;
    const int  blk  = bt_row[cc >> 8];                          // uniform -> s_load
    const long slot = (long)blk * BLOCK_SIZE + (cc & (BLOCK_SIZE - 1));
    const long eoff = slot * (NUM_KV_HEADS * HEAD_DIM) + (long)kh * HEAD_DIM;
    tdm_load_tile(kc + eoff, (unsigned)((w * 2 + buf) * TILE_F * 4));
    tdm_load_tile(vc + eoff, (unsigned)((8 + w * 2 + buf) * TILE_F * 4));
}

__global__ __launch_bounds__(32 * WAVES)
void paged_attn_decode(const float* __restrict__ q,
                       const float* __restrict__ knew,
                       const float* __restrict__ vnew,
                       const float* __restrict__ kc,
                       const float* __restrict__ vc,
                       const int* __restrict__ block_table,
                       const int* __restrict__ ctx_lens,
                       float* __restrict__ out)
{
    const int kh   = blockIdx.x;
    const int b    = blockIdx.y;
    const int lane = threadIdx.x & 31;
    const int w    = threadIdx.x >> 5;
    const int n    = lane & 15;   // matrix column (query for scores, dim for V)
    const int h    = lane >> 4;   // half-wave selector (K-pair split)

    const int ctx    = ctx_lens[b];
    const int nchunk = (ctx + 15) >> 4;
    const int* bt_row = block_table + b * MAX_BLOCKS;

    // ---- preload Q as 32 B-fragments (4x16, K-dim groups of 4) ----
    const float* qrow = q + (size_t)b * NUM_HEADS * HEAD_DIM
                          + (size_t)(kh * GROUP + (n & 3)) * HEAD_DIM;
    v2f qf[32];
#pragma unroll
    for (int j = 0; j < 32; ++j) {
        if (n < GROUP) {
            qf[j].x = qrow[4 * j + 2 * h];
            qf[j].y = qrow[4 * j + 2 * h + 1];
        } else {
            qf[j].x = 0.f; qf[j].y = 0.f;
        }
    }

    const float* knew_b = knew + (size_t)b * NUM_KV_HEADS * HEAD_DIM + (size_t)kh * HEAD_DIM;
    const float* vnew_b = vnew + (size_t)b * NUM_KV_HEADS * HEAD_DIM + (size_t)kh * HEAD_DIM;

    v8f acc[8];
#pragma unroll
    for (int t = 0; t < 8; ++t) acc[t] = (v8f){0.f,0.f,0.f,0.f,0.f,0.f,0.f,0.f};
    float m_run = -1e30f;
    float l_run = 0.f;
    const float scale = 0.08838834764831845f;  // 1/sqrt(128)

    // prime the double buffer
    if (w < nchunk) prefetch_chunk(kc, vc, bt_row, kh, w, w, 0);

    int buf = 0;
    for (int c = w; c < nchunk; c += WAVES, buf ^= 1) {
        const int sbase = c * 16;

        // prefetch next chunk into the other buffer, then land the current one
        if (c + WAVES < nchunk) {
            prefetch_chunk(kc, vc, bt_row, kh, c + WAVES, w, buf ^ 1);
            __builtin_amdgcn_s_wait_tensorcnt(2);
        } else {
            __builtin_amdgcn_s_wait_tensorcnt(0);
        }
        __asm__ volatile("" ::: "memory");

        float* kt = smem + (w * 2 + buf) * TILE_F;
        float* vt = smem + (8 + w * 2 + buf) * TILE_F;

        // patch the newly appended token's row (position ctx-1) over the cache data
        if ((sbase >> 4) == ((ctx - 1) >> 4)) {
            const int r = (ctx - 1) & 15;
            *(float4*)(kt + r * ROW_F + lane * 4) = *(const float4*)(knew_b + lane * 4);
            *(float4*)(vt + r * ROW_F + lane * 4) = *(const float4*)(vnew_b + lane * 4);
        }

        // ---- scores^T = K_tile x Q^T : 32 WMMAs over the 128-dim ----
        // Double-buffered register fragments (groups of 4) + two independent
        // accumulator chains so DS latency and XDL latency overlap.
        const float* kbase = kt + n * ROW_F + 2 * h;   // this lane's token row
        v2f ka[2][4];
#pragma unroll
        for (int jj = 0; jj < 4; ++jj) ka[0][jj] = *(const v2f*)(kbase + 4 * jj);

        v8f sc0 = (v8f){0.f,0.f,0.f,0.f,0.f,0.f,0.f,0.f};
        v8f sc1 = (v8f){0.f,0.f,0.f,0.f,0.f,0.f,0.f,0.f};
#pragma unroll
        for (int g = 0; g < 8; ++g) {
            const int cb = g & 1;
            if (g < 7) {
#pragma unroll
                for (int jj = 0; jj < 4; ++jj)
                    ka[cb ^ 1][jj] = *(const v2f*)(kbase + 4 * (4 * g + 4 + jj));
            }
#pragma unroll
            for (int jj = 0; jj < 4; ++jj) {
                if (jj & 1)
                    sc1 = __builtin_amdgcn_wmma_f32_16x16x4_f32(
                              false, ka[cb][jj], false, qf[4 * g + jj], (short)0, sc1, false, false);
                else
                    sc0 = __builtin_amdgcn_wmma_f32_16x16x4_f32(
                              false, ka[cb][jj], false, qf[4 * g + jj], (short)0, sc0, false, false);
            }
        }

        // ---- scale + length mask, chunk max ----
        float preg[8];
        float rowmax = -1e30f;
#pragma unroll
        for (int v = 0; v < 8; ++v) {
            const int tok = sbase + v + 8 * h;
            float s = (sc0[v] + sc1[v]) * scale;
            if (tok >= ctx) s = -1e30f;
            preg[v] = s;
            rowmax  = fmaxf(rowmax, s);
        }
        rowmax = fmaxf(rowmax, __shfl_xor(rowmax, 16, 32));

        const float m_new = fmaxf(m_run, rowmax);
        const float corr  = __expf(m_run - m_new);

        float rowsum = 0.f;
#pragma unroll
        for (int v = 0; v < 8; ++v) {
            const float p = __expf(preg[v] - m_new);
            preg[v] = p;
            rowsum += p;
        }
        rowsum += __shfl_xor(rowsum, 16, 32);
        l_run = l_run * corr + rowsum;
        m_run = m_new;

        // rescale V accumulators: per-query correction lives in lanes 0..3
        float scv[4];
        scv[0] = __shfl(corr, 0, 32);
        scv[1] = __shfl(corr, 1, 32);
        scv[2] = __shfl(corr, 2, 32);
        scv[3] = __shfl(corr, 3, 32);
#pragma unroll
        for (int t = 0; t < 8; ++t)
#pragma unroll
            for (int v = 0; v < 8; ++v)
                acc[t][v] *= scv[v & 3];

        // partner-half p values for A-fragment assembly
        float pswap[8];
#pragma unroll
        for (int v = 0; v < 8; ++v) pswap[v] = __shfl_xor(preg[v], 16, 32);

        // ---- out += P^T x V : 4 token-subgroups x 8 dim tiles, pipelined ----
        v2f vfb[2][8];
        {
            const float* vr0 = vt + (2 * h) * ROW_F;
            const float* vr1 = vr0 + ROW_F;
#pragma unroll
            for (int t = 0; t < 8; ++t) {
                vfb[0][t].x = vr0[16 * t + n];
                vfb[0][t].y = vr1[16 * t + n];
            }
        }
#pragma unroll
        for (int sg = 0; sg < 4; ++sg) {
            const int cb = sg & 1;
            if (sg < 3) {
                const float* vr0 = vt + (4 * (sg + 1) + 2 * h) * ROW_F;
                const float* vr1 = vr0 + ROW_F;
#pragma unroll
                for (int t = 0; t < 8; ++t) {
                    vfb[cb ^ 1][t].x = vr0[16 * t + n];
                    vfb[cb ^ 1][t].y = vr1[16 * t + n];
                }
            }
            const int  t0  = 4 * sg + 2 * h;         // this lane's first token in subgroup
            const bool own = ((t0 >> 3) == h);       // token lives in own half-wave regs
            v2f pa;
            pa.x = own ? preg[t0 & 7]       : pswap[t0 & 7];
            pa.y = own ? preg[(t0 & 7) + 1] : pswap[(t0 & 7) + 1];
#pragma unroll
            for (int t = 0; t < 8; ++t)
                acc[t] = __builtin_amdgcn_wmma_f32_16x16x4_f32(
                             false, pa, false, vfb[cb][t], (short)0, acc[t], false, false);
        }
    }

    // ---- per-wave partials to LDS ----
    float* s_acc = smem + MERGE_F;                       // [WAVES][GROUP][HEAD_DIM]
    float* s_m   = s_acc + WAVES * GROUP * HEAD_DIM;     // [WAVES][GROUP]
    float* s_l   = s_m + WAVES * GROUP;                  // [WAVES][GROUP]

    if (lane < 16) {
#pragma unroll
        for (int v = 0; v < GROUP; ++v)
#pragma unroll
            for (int t = 0; t < 8; ++t)
                s_acc[(w * GROUP + v) * HEAD_DIM + 16 * t + lane] = acc[t][v];
    }
    if (lane < GROUP) {   // lane v already holds query v's state
        s_m[w * GROUP + lane] = m_run;
        s_l[w * GROUP + lane] = l_run;
    }
    __syncthreads();

    // ---- cross-wave flash-decode merge: 512 outputs over 128 threads ----
#pragma unroll
    for (int i = 0; i < 4; ++i) {
        const int idx = threadIdx.x + 128 * i;
        const int qv  = idx >> 7;
        const int d   = idx & 127;
        float mstar = -1e30f;
#pragma unroll
        for (int ww = 0; ww < WAVES; ++ww) mstar = fmaxf(mstar, s_m[ww * GROUP + qv]);
        float lsum = 0.f, osum = 0.f;
#pragma unroll
        for (int ww = 0; ww < WAVES; ++ww) {
            const float e = __expf(s_m[ww * GROUP + qv] - mstar);
            lsum += e * s_l[ww * GROUP + qv];
            osum += e * s_acc[(ww * GROUP + qv) * HEAD_DIM + d];
        }
        out[(size_t)b * NUM_HEADS * HEAD_DIM + (size_t)(kh * GROUP + qv) * HEAD_DIM + d] =
            osum / lsum;
    }
}

extern "C" void kernel_launch(void* const* d_in, const int* in_sizes, int n_in,
                              void* d_out, int out_size, void* d_ws, size_t ws_size,
                              hipStream_t stream) {
    const float* q   = (const float*)d_in[0];
    const float* k   = (const float*)d_in[1];
    const float* v   = (const float*)d_in[2];
    const float* kc  = (const float*)d_in[3];
    const float* vc  = (const float*)d_in[4];
    const int*   bt  = (const int*)d_in[5];
    const int*   cl  = (const int*)d_in[6];
    // d_in[7] slot_mapping not needed: slot == position ctx-1 by construction

    const int B = in_sizes[6];  // context_lens has one entry per sequence
    dim3 grid(NUM_KV_HEADS, B);
    paged_attn_decode<<<grid, 32 * WAVES, LDS_BYTES, stream>>>(q, k, v, kc, vc, bt, cl,
                                                              (float*)d_out);
}